// HardUpsampling_14078902796990
// MI455X (gfx1250) — compile-verified
//
#include <hip/hip_runtime.h>

#define N_BATCH 16
#define T_SEQ   2048
#define M_DIM   256

typedef float v4f __attribute__((ext_vector_type(4)));  // native 16B vector

// ---------------------------------------------------------------------------
// Kernel 1: per-batch inclusive prefix sum of durations -> cum[] (workspace),
//           total length -> lens tail of d_out (as float, output dtype f32).
// One block per batch, 256 threads, 8 durations per thread.
// ---------------------------------------------------------------------------
__global__ void upsample_scan_kernel(const int* __restrict__ durations,
                                     int* __restrict__ cum,
                                     float* __restrict__ lens_out) {
  const int n = blockIdx.x;
  const int t = threadIdx.x;                // 0..255
  const int* dur  = durations + n * T_SEQ;
  int*       cumn = cum       + n * T_SEQ;

  // Local inclusive scan of 8 consecutive durations.
  int v[8];
  const int base = t * 8;
  int s = 0;
#pragma unroll
  for (int k = 0; k < 8; ++k) { s += dur[base + k]; v[k] = s; }

  // Block-wide Hillis-Steele inclusive scan over the 256 per-thread sums.
  __shared__ int part[256];
  part[t] = s;
  __syncthreads();
  for (int off = 1; off < 256; off <<= 1) {
    int add = (t >= off) ? part[t - off] : 0;
    __syncthreads();
    part[t] += add;
    __syncthreads();
  }
  const int excl = part[t] - s;             // exclusive prefix for this thread
#pragma unroll
  for (int k = 0; k < 8; ++k) cumn[base + k] = excl + v[k];

  if (t == 255) lens_out[n] = (float)part[255];
}

// ---------------------------------------------------------------------------
// Kernel 2: gather/upsample. Block = 256 threads (8 waves), handles 32 output
// rows of one batch. cum[n][0..2047] (8KB) is staged into LDS with gfx1250
// async global->LDS b128 copies (ASYNCcnt), then each wave binary-searches in
// LDS and streams one 1KB row per iteration with 128-bit accesses.
// Output stores are non-temporal (th:TH_STORE_NT): written once, never
// re-read -> keep the 192MB L2 for the 32MB seqs gather working set.
// ---------------------------------------------------------------------------
__global__ void upsample_gather_kernel(const float* __restrict__ seqs,
                                       const int* __restrict__ cum,
                                       float* __restrict__ out,
                                       int L) {
  extern __shared__ int cum_s[];            // T_SEQ ints = 8KB, dyn-LDS base 0
  const int n    = blockIdx.y;
  const int tid  = (int)threadIdx.x;        // 0..255
  const int lane = tid & 31;
  const int wave = tid >> 5;

  // ---- async stage cum[n] -> LDS: 256 lanes x 2 x 16B = 8192B -------------
  {
    const int* g = cum + n * T_SEQ;
#pragma unroll
    for (int it = 0; it < 2; ++it) {
      unsigned   lds_off = (unsigned)(tid * 16 + it * 4096); // byte offset in LDS
      const int* gaddr   = g + tid * 4 + it * 1024;          // 4 ints per lane
      asm volatile("global_load_async_to_lds_b128 %0, %1, off"
                   :: "v"(lds_off), "v"(gaddr)
                   : "memory");
    }
    asm volatile("s_wait_asynccnt 0x0" ::: "memory");
  }
  __syncthreads();

  const int len = cum_s[T_SEQ - 1];         // upsampled length for this batch

#pragma unroll
  for (int r = 0; r < 4; ++r) {
    const int l = blockIdx.x * 32 + wave * 4 + r;   // output row (position)
    if (l >= L) continue;

    v4f a, b;
    if (l < len) {
      // searchsorted(cum, l, side='right'): first j with cum_s[j] > l.
      // Uniform across the wave -> scalar-friendly LDS broadcast reads.
      int lo = 0, hi = T_SEQ;
      while (lo < hi) {
        const int mid = (lo + hi) >> 1;
        if (cum_s[mid] <= l) lo = mid + 1; else hi = mid;
      }
      const int idx = lo < T_SEQ ? lo : (T_SEQ - 1);
      const v4f* src =
          (const v4f*)(seqs + ((size_t)n * T_SEQ + idx) * (size_t)M_DIM);
      a = src[lane * 2 + 0];                // 32 lanes x 32B = 1KB row, coalesced
      b = src[lane * 2 + 1];
    } else {
      a = (v4f){0.f, 0.f, 0.f, 0.f};        // padded slot -> zeros
      b = a;
    }
    v4f* dst = (v4f*)(out + ((size_t)n * L + l) * (size_t)M_DIM);
    __builtin_nontemporal_store(a, &dst[lane * 2 + 0]);   // th:TH_STORE_NT
    __builtin_nontemporal_store(b, &dst[lane * 2 + 1]);
  }
}

// ---------------------------------------------------------------------------
extern "C" void kernel_launch(void* const* d_in, const int* in_sizes, int n_in,
                              void* d_out, int out_size, void* d_ws, size_t ws_size,
                              hipStream_t stream) {
  (void)in_sizes; (void)n_in; (void)ws_size;

  const float* seqs      = (const float*)d_in[0];   // [16, 2048, 256] f32
  const int*   durations = (const int*)d_in[1];     // [16, 2048] i32
  float*       out       = (float*)d_out;

  const int N = N_BATCH, M = M_DIM;
  const int L = (out_size - N) / (N * M);           // padded output length

  int*   cum      = (int*)d_ws;                     // [16, 2048] i32 scratch
  float* lens_out = out + (size_t)out_size - N;     // lens appended after upsampled

  upsample_scan_kernel<<<dim3(N), dim3(256), 0, stream>>>(durations, cum, lens_out);

  dim3 grid((L + 31) / 32, N);
  upsample_gather_kernel<<<grid, dim3(256), T_SEQ * sizeof(int), stream>>>(
      seqs, cum, out, L);
}